// BARS_76733885710679
// MI455X (gfx1250) — compile-verified
//
#include <hip/hip_runtime.h>
#include <hip/hip_bf16.h>

// CDNA5 / gfx1250, wave32.
typedef __attribute__((ext_vector_type(2))) float v2f;
typedef __attribute__((ext_vector_type(4))) float v4f;
typedef __attribute__((ext_vector_type(8))) float v8f;

#define B_    4
#define C_    256
#define HF    64
#define WF    128
#define K_    19
#define HL    512
#define WL    1024
#define HW    (HF * WF)          // 8192
#define WAVES 4
#define THREADS (WAVES * 32)
#define BLOCKS_PER_TENSOR ((B_ * HF * (WF / 16)) / WAVES)   // 4*64*8/4 = 512

__global__ __launch_bounds__(THREADS)
void BARS_centroid_mask_kernel(const float* __restrict__ f0,
                               const float* __restrict__ f1,
                               const float* __restrict__ cen_s2t,
                               const float* __restrict__ cen_tgt,
                               float* __restrict__ out)
{
    // Centroid B-matrices pre-swizzled into the exact WMMA per-lane layout:
    // sB[s][lane][j] = B(k = 4s + j + 2*(lane>>4), n), n = lane&15 (+16 for tile 1)
    __shared__ float sB0[64][32][2];              // 16 KB
    __shared__ float sB1[64][32][2];              // 16 KB
    __shared__ float sC2[32];                     // centroid squared norms
    __shared__ float sScore[WAVES][16][32];       // 8 KB  (D tiles, [pixel][centroid])
    __shared__ int   sMask[WAVES][16];

    const int t   = blockIdx.x / BLOCKS_PER_TENSOR;   // 0: s2t feats, 1: target feats
    const int blk = blockIdx.x % BLOCKS_PER_TENSOR;
    const float* feat = (t == 0) ? f0 : f1;
    // CROSS assignment: s2t features vs TARGET centroids, and vice versa.
    const float* cen  = (t == 0) ? cen_tgt : cen_s2t;

    const int tid = threadIdx.x;

    // ---- Stage centroids into LDS in WMMA B layout ----
    for (int idx = tid; idx < 64 * 32 * 2; idx += THREADS) {
        int j = idx & 1;
        int l = (idx >> 1) & 31;
        int s = idx >> 6;
        int kc = 4 * s + j + 2 * (l >> 4);
        int n0 = l & 15;
        sB0[s][l][j] = cen[n0 * C_ + kc];
        int n1 = 16 + n0;
        sB1[s][l][j] = (n1 < K_) ? cen[n1 * C_ + kc] : 0.0f;
    }
    if (tid < 32) {
        float acc = 0.0f;
        if (tid < K_) {
            #pragma unroll 8
            for (int c = 0; c < C_; ++c) { float v = cen[tid * C_ + c]; acc += v * v; }
        }
        sC2[tid] = acc;
    }
    __syncthreads();

    const int wave  = tid >> 5;
    const int lane  = tid & 31;
    const int m     = lane & 15;    // pixel within tile (A rows) == column n (C/D)
    const int khalf = lane >> 4;

    // tile -> (b, h, 16-pixel group along W)
    const int tileIdx = blk * WAVES + wave;
    const int wt = tileIdx & 7;
    const int bh = tileIdx >> 3;
    const int h  = bh & (HF - 1);
    const int b  = bh >> 6;
    const int pw = wt * 16 + m;

    // This lane's pixel: channel c lives at fp[c * HW]
    const float* fp = feat + (size_t)b * C_ * HW + (size_t)h * WF + pw;

    // Init accumulators with -0.5*||c_n||^2 (padded columns -> huge negative).
    const float c20 = -0.5f * sC2[m];
    const float c21 = (16 + m < K_) ? (-0.5f * sC2[16 + m]) : -1.0e30f;
    v8f acc0, acc1;
    #pragma unroll
    for (int r = 0; r < 8; ++r) { acc0[r] = c20; acc1[r] = c21; }

    // ---- K loop: 64 steps of 4 channels, 2 WMMAs per step ----
    #pragma unroll 4
    for (int s = 0; s < 64; ++s) {
        const int kc = 4 * s + 2 * khalf;           // A: VGPR0=K kc, VGPR1=K kc+1
        v2f a;
        a.x = fp[(size_t)kc * HW];
        a.y = fp[(size_t)(kc + 1) * HW];
        v2f b0 = *(const v2f*)&sB0[s][lane][0];
        v2f b1 = *(const v2f*)&sB1[s][lane][0];
        acc0 = __builtin_amdgcn_wmma_f32_16x16x4_f32(
            false, a, false, b0, (short)0, acc0, false, false);
        acc1 = __builtin_amdgcn_wmma_f32_16x16x4_f32(
            false, a, false, b1, (short)0, acc1, false, false);
    }

    // ---- Spill D tiles to LDS: row M = r + 8*half, col N = lane&15 ----
    #pragma unroll
    for (int r = 0; r < 8; ++r) {
        int row = r + 8 * khalf;
        sScore[wave][row][m]      = acc0[r];
        sScore[wave][row][16 + m] = acc1[r];
    }
    __syncthreads();

    // ---- Argmax per pixel (ascending scan, strict '>' => first-index ties) ----
    if (lane < 16) {
        float best = sScore[wave][lane][0];
        int bi = 0;
        #pragma unroll
        for (int n = 1; n < K_; ++n) {
            float v = sScore[wave][lane][n];
            if (v > best) { best = v; bi = n; }
        }
        sMask[wave][lane] = bi;
    }
    __syncthreads();

    // ---- Fused 8x8 nearest upsample: lane halves split the 8 output rows ----
    const float mv = (float)sMask[wave][m];
    const v4f mv4 = { mv, mv, mv, mv };
    const size_t outBase = ((size_t)t * B_ + b) * ((size_t)HL * WL);
    const int orow0 = 8 * h + 4 * khalf;
    const int ocol  = 8 * pw;
    #pragma unroll
    for (int r = 0; r < 4; ++r) {
        float* p = out + outBase + (size_t)(orow0 + r) * WL + ocol;
        *(v4f*)p       = mv4;    // global_store_b128
        *(v4f*)(p + 4) = mv4;    // global_store_b128
    }
}

extern "C" void kernel_launch(void* const* d_in, const int* in_sizes, int n_in,
                              void* d_out, int out_size, void* d_ws, size_t ws_size,
                              hipStream_t stream) {
    (void)in_sizes; (void)n_in; (void)out_size; (void)d_ws; (void)ws_size;
    const float* f_s2t  = (const float*)d_in[0];  // feature_s2t  [4,256,64,128]
    const float* f_tgt  = (const float*)d_in[1];  // feature_target
    // d_in[2], d_in[3] labels: unused (output shape is static)
    const float* c_s2t  = (const float*)d_in[4];  // centroid_s2t [19,256]
    const float* c_tgt  = (const float*)d_in[5];  // centroid_target
    float* out = (float*)d_out;                   // [2, 4, 512, 1024] class ids

    dim3 grid(2 * BLOCKS_PER_TENSOR);
    dim3 block(THREADS);
    BARS_centroid_mask_kernel<<<grid, block, 0, stream>>>(f_s2t, f_tgt, c_s2t, c_tgt, out);
}